// SGRUCell_11768210391739
// MI455X (gfx1250) — compile-verified
//
#include <hip/hip_runtime.h>
#include <math.h>

// Problem constants (from reference): B=4096, H=256, I=32, 3H=768.
#define B_N   4096
#define H_N   256
#define I_N   32
#define G3    768                 // 3*H
#define GH_N  (G3 * H_N)          // 196608 = elements of one expert slice

typedef float v2f __attribute__((ext_vector_type(2)));
typedef float v8f __attribute__((ext_vector_type(8)));

// ---------------------------------------------------------------------------
// Workspace layout (d_ws):
//   [0,   128)                      int   counts[32]
//   [256, 256+512KB)                int   bucket[32][4096]
//   [524544, +25.2MB)               float W_t[32][768][256]   (h contiguous)
// Total ≈ 24.5 MB. If ws_size is smaller, we fall back to gathering the
// B-matrix straight from the original (stride-32-float) layout.
// ---------------------------------------------------------------------------

__global__ __launch_bounds__(32)
void zero_counts_kernel(int* counts) {
    if (threadIdx.x < I_N) counts[threadIdx.x] = 0;
}

__global__ __launch_bounds__(256)
void bucket_build_kernel(const float* __restrict__ inp,
                         int* __restrict__ counts,
                         int* __restrict__ bucket) {
    int b = blockIdx.x * blockDim.x + threadIdx.x;
    if (b >= B_N) return;
    // input is one-hot fp32: argmax over 32 entries
    int   idx  = 0;
    float best = -1.0f;
#pragma unroll
    for (int i = 0; i < I_N; ++i) {
        float v = inp[b * I_N + i];
        if (v > best) { best = v; idx = i; }
    }
    int pos = atomicAdd(&counts[idx], 1);
    bucket[idx * B_N + pos] = b;
}

// Transpose weight_hh [GH_N rows x 32 experts] -> W_t [32][GH_N].
// Original flat index: g*8192 + h*32 + e  => row (g*256+h), col e.
// Output W_t[e][g*256+h]  (i.e. [e][g][h], h contiguous).
__global__ __launch_bounds__(256)
void transpose_w_kernel(const float* __restrict__ w, float* __restrict__ wt) {
    __shared__ float tile[64][33];          // +1 pad: conflict-free both ways
    const int rbase = blockIdx.x * 64;
    const int t = threadIdx.x;
#pragma unroll
    for (int p = 0; p < 8; ++p) {           // coalesced reads: contiguous flat
        int elem = t + p * 256;
        int r = elem >> 5;                  // 0..63
        int c = elem & 31;                  // expert
        tile[r][c] = w[(size_t)(rbase + r) * I_N + c];
    }
    __syncthreads();
#pragma unroll
    for (int p = 0; p < 8; ++p) {           // coalesced writes: 64-wide runs
        int elem = t + p * 256;
        int c = elem >> 6;                  // expert 0..31
        int r = elem & 63;
        wt[(size_t)c * GH_N + rbase + r] = tile[r][c];
    }
}

// ---------------------------------------------------------------------------
// Per-(expert, 16-row tile) GEMM + GRU gates.
//   rec[m, g] = sum_h  B[h][g] * A[m][h]
//   A = hx rows (staged in LDS), B[h][g] = weight slice of expert e.
// 256 threads = 8 wave32s; wave w owns N-range [w*96, w*96+96) (6 N-tiles).
// K = 256 streamed in steps of 4 with V_WMMA_F32_16X16X4_F32.
// ---------------------------------------------------------------------------
#define A_RS   258   // LDS row stride for A  (even, bank-conflict friendly)
#define REC_RS 772   // LDS row stride for rec (8*772 % 64 == 32 -> halves split)

template <bool TRANSPOSED>
__global__ __launch_bounds__(256)
void gemm_gru_kernel(const float* __restrict__ hx,
                     const float* __restrict__ bias_ih,
                     const float* __restrict__ bias_hh,
                     const float* __restrict__ w_orig,   // [768][256][32]
                     const float* __restrict__ wt,       // [32][768][256]
                     const int*   __restrict__ counts,
                     const int*   __restrict__ bucket,
                     float* __restrict__ out) {
    const int e   = blockIdx.y;
    const int m0  = blockIdx.x * 16;
    const int cnt = counts[e];
    if (m0 >= cnt) return;                  // uniform exit, no barrier reached

    __shared__ float lds_A[16 * A_RS];      // 16 x 256 hx tile
    __shared__ float lds_rec[16 * REC_RS];  // 16 x 768 rec tile
    __shared__ int   rows[16];

    const int tid = threadIdx.x;
    if (tid < 16) {
        int mi = m0 + tid;
        rows[tid] = (mi < cnt) ? bucket[e * B_N + mi] : -1;
    }
    __syncthreads();

    // Stage A (selected hx rows) into LDS; zero-fill invalid rows.
    {
        int r  = tid >> 4;
        int c0 = tid & 15;
        int rb = rows[r];
#pragma unroll
        for (int j = 0; j < 16; ++j) {
            int c = c0 + j * 16;
            lds_A[r * A_RS + c] = (rb >= 0) ? hx[(size_t)rb * H_N + c] : 0.0f;
        }
    }
    __syncthreads();

    const int lane = tid & 31;
    const int wave = tid >> 5;
    const int mh   = lane & 15;             // M for A-frag / N for B-frag
    const int koff = (lane >> 4) << 1;      // lanes 0-15: K+{0,1}; 16-31: K+{2,3}

    const float* We = wt + (size_t)e * GH_N;

    v8f acc[6];
    const v8f vzero = {0.f, 0.f, 0.f, 0.f, 0.f, 0.f, 0.f, 0.f};
#pragma unroll
    for (int t = 0; t < 6; ++t) acc[t] = vzero;

    for (int k = 0; k < H_N; k += 4) {
        // A frag: A[mh][k+koff], A[mh][k+koff+1]  (ds b64)
        v2f a = *(const v2f*)&lds_A[mh * A_RS + k + koff];
#pragma unroll
        for (int t = 0; t < 6; ++t) {
            int n = wave * 96 + t * 16 + mh;
            v2f b;
            if (TRANSPOSED) {
                // B[h][n] = W_t[e][n*256 + h] : contiguous b64 per lane
                b = *(const v2f*)&We[(size_t)n * H_N + k + koff];
            } else {
                // gather from original layout: w[n*8192 + h*32 + e]
                const float* p = w_orig + (size_t)n * (H_N * I_N)
                               + (size_t)(k + koff) * I_N + e;
                b.x = p[0];
                b.y = p[I_N];
            }
            acc[t] = __builtin_amdgcn_wmma_f32_16x16x4_f32(
                false, a, false, b, (short)0, acc[t], false, false);
        }
    }

    // D layout: vgpr j, lane L -> (M = j + 8*(L>=16), N = L&15)
#pragma unroll
    for (int t = 0; t < 6; ++t) {
        int n = wave * 96 + t * 16 + mh;
#pragma unroll
        for (int j = 0; j < 8; ++j) {
            lds_rec[(j + ((lane >> 4) << 3)) * REC_RS + n] = acc[t][j];
        }
    }
    __syncthreads();

    // Gate math: thread handles (row m = tid>>4, 16 h values), coalesced in h.
    {
        int m  = tid >> 4;
        int rb = rows[m];
        if (rb >= 0) {
#pragma unroll
            for (int j = 0; j < 16; ++j) {
                int h = (tid & 15) + j * 16;
                float ri = bias_ih[(size_t)h * I_N + e];
                float ui = bias_ih[(size_t)(h + H_N) * I_N + e];
                float ni = bias_ih[(size_t)(h + 2 * H_N) * I_N + e];
                float rh = lds_rec[m * REC_RS + h]
                         + bias_hh[(size_t)h * I_N + e];
                float uh = lds_rec[m * REC_RS + h + H_N]
                         + bias_hh[(size_t)(h + H_N) * I_N + e];
                float nh = lds_rec[m * REC_RS + h + 2 * H_N]
                         + bias_hh[(size_t)(h + 2 * H_N) * I_N + e];
                float rg = 1.0f / (1.0f + __expf(-(ri + rh)));
                float ug = 1.0f / (1.0f + __expf(-(ui + uh)));
                float ng = tanhf(ni + rg * nh);
                float hv = hx[(size_t)rb * H_N + h];
                out[(size_t)rb * H_N + h] = ug * hv + (1.0f - ug) * ng;
            }
        }
    }
}

extern "C" void kernel_launch(void* const* d_in, const int* in_sizes, int n_in,
                              void* d_out, int out_size, void* d_ws, size_t ws_size,
                              hipStream_t stream) {
    const float* inp = (const float*)d_in[0];   // [B, I]  one-hot
    const float* hx  = (const float*)d_in[1];   // [B, H]
    const float* w   = (const float*)d_in[2];   // [3H, H, I]
    const float* bih = (const float*)d_in[3];   // [3H, I]
    const float* bhh = (const float*)d_in[4];   // [3H, I]
    float* out = (float*)d_out;                 // [1, B, H]

    int*   counts = (int*)d_ws;
    int*   bucket = (int*)((char*)d_ws + 256);
    float* wt     = (float*)((char*)d_ws + 256 + (size_t)I_N * B_N * sizeof(int));
    const size_t need = 256 + (size_t)I_N * B_N * sizeof(int)
                      + (size_t)I_N * GH_N * sizeof(float);

    zero_counts_kernel<<<1, 32, 0, stream>>>(counts);
    bucket_build_kernel<<<B_N / 256, 256, 0, stream>>>(inp, counts, bucket);

    dim3 grid(B_N / 16, I_N);
    if (ws_size >= need) {
        transpose_w_kernel<<<GH_N / 64, 256, 0, stream>>>(w, wt);
        gemm_gru_kernel<true><<<grid, 256, 0, stream>>>(
            hx, bih, bhh, w, wt, counts, bucket, out);
    } else {
        gemm_gru_kernel<false><<<grid, 256, 0, stream>>>(
            hx, bih, bhh, w, wt, counts, bucket, out);
    }
}